// PropGTNet_30133490549596
// MI455X (gfx1250) — compile-verified
//
#include <hip/hip_runtime.h>
#include <math.h>

typedef float v2f __attribute__((ext_vector_type(2)));
typedef float v8f __attribute__((ext_vector_type(8)));

#define BN_EPS 1e-5f
#define KL_EPS 1e-16f

// ---- native fp32 WMMA: D(16x16) = A(16x4) * B(4x16) + C ----
__device__ __forceinline__ v8f wmma4(v2f a, v2f b, v8f c) {
  return __builtin_amdgcn_wmma_f32_16x16x4_f32(false, a, false, b, (short)0, c,
                                               false, false);
}

// ---------------- workspace layout (float offsets) ----------------
static constexpr int REL_SUM    = 0;         // 256
static constexpr int REL_SSQ    = 256;
static constexpr int WGT_SUM    = 512;
static constexpr int WGT_SSQ    = 768;
static constexpr int SELF_SUM   = 1024;
static constexpr int SELF_SSQ   = 1280;
static constexpr int PRED_SUM   = 1536;
static constexpr int PRED_SSQ   = 1792;
static constexpr int REL_SCALE  = 2048;      // 256
static constexpr int REL_BIAS2  = 2304;      // 128
static constexpr int WGT_SCALE  = 2432;      // 256
static constexpr int WGT_BIAS2  = 2688;      // 2 (pad 16)
static constexpr int SELF_SCALE = 2704;      // 256
static constexpr int SELF_BIAS2 = 2960;      // 128
static constexpr int PRED_SCALE = 3088;      // 256
static constexpr int PRED_BIAS2 = 3344;      // 2 (pad 16)
static constexpr int KL_ACC     = 3360;      // 1
static constexpr int STATS_LEN  = 4096;
static constexpr int H_SELF     = 4096;                    // 4096x256
static constexpr int XBUF       = H_SELF + 4096 * 256;     // 4096x256 (enc_self|enc_rel)
static constexpr int GI_BUF     = XBUF + 4096 * 256;       // 4096x384
static constexpr int GH_BUF     = GI_BUF + 4096 * 384;     // 4096x384
static constexpr int H_PRED     = GH_BUF + 4096 * 384;     // 4096x256

// d_out layout (floats)
static constexpr int OUT_UPD = 0;            // 64*64*128
static constexpr int OUT_ZW  = 524288;       // 64*128
static constexpr int OUT_W   = 532480;       // 64*64*64
static constexpr int OUT_KL  = 794624;       // 1

// ---------------- zero ----------------
__global__ void k_zero(float* __restrict__ p, int n) {
  int i = blockIdx.x * 256 + threadIdx.x;
  if (i < n) p[i] = 0.f;
}

// ---------------- self MLP hidden (one-hot collapse) ----------------
// feat = [z_where(2) | eye(64) | actions(2)];  h = relu(feat @ w1 + b1)
__global__ void __launch_bounds__(256)
k_self_hidden(const float* __restrict__ z_where, const float* __restrict__ actions,
              const float* __restrict__ w1, const float* __restrict__ b1,
              float* __restrict__ h, float* __restrict__ sum, float* __restrict__ ssq) {
  int t = threadIdx.x;                       // hidden column
  int r0 = blockIdx.x * 16;                  // 16 rows per block
  float w0  = w1[0 * 256 + t];
  float w1c = w1[1 * 256 + t];
  float wa0 = w1[66 * 256 + t];
  float wa1 = w1[67 * 256 + t];
  float bc  = b1[t];
  float lsum = 0.f, lssq = 0.f;
  #pragma unroll 4
  for (int rr = 0; rr < 16; ++rr) {
    int row = r0 + rr;
    int b = row >> 6, n = row & 63;
    float zw0 = z_where[row * 2 + 0], zw1 = z_where[row * 2 + 1];
    float a0 = actions[b * 2 + 0], a1 = actions[b * 2 + 1];
    float hv = zw0 * w0 + zw1 * w1c + w1[(2 + n) * 256 + t] + a0 * wa0 + a1 * wa1 + bc;
    hv = fmaxf(hv, 0.f);
    h[row * 256 + t] = hv;
    lsum += hv; lssq += hv * hv;
  }
  atomicAdd(&sum[t], lsum);
  atomicAdd(&ssq[t], lssq);
}

// ---------------- BN fold finalize ----------------
// scale[k] = g[k]*rsqrt(var+eps);  bias2[n] = b2[n] + sum_k (beta[k]-mu[k]*scale[k])*w2[k][n]
__global__ void __launch_bounds__(256)
k_finalize(const float* __restrict__ sum, const float* __restrict__ ssq, float count,
           const float* __restrict__ g, const float* __restrict__ beta,
           const float* __restrict__ w2, int ldb, const float* __restrict__ b2, int N2,
           float* __restrict__ scale, float* __restrict__ bias2) {
  __shared__ float shift[256];
  int t = threadIdx.x;
  float mu = sum[t] / count;
  float var = ssq[t] / count - mu * mu;
  float s = g[t] * rsqrtf(var + BN_EPS);
  scale[t] = s;
  shift[t] = beta[t] - mu * s;
  __syncthreads();
  if (t < N2) {
    float acc = b2[t];
    #pragma unroll 8
    for (int k = 0; k < 256; ++k) acc += shift[k] * w2[k * ldb + t];
    bias2[t] = acc;
  }
}

// ---------------- generic WMMA GEMM (templated: zero runtime branches) ------
// C[m,n] = relu?( sum_k A[m,k] * (scale[k]*B[k,n]) + bias[n] )
// block = 256 thr = 8 waves -> 32(M) x 64(N) tile; grid = (M/32, N/64)
// Requires: N % 64 == 0, K % 64 == 0 (all call sites satisfy this).
template <bool HAS_SCALE, bool RELU, bool STATS>
__global__ void __launch_bounds__(256)
k_gemm(const float* __restrict__ A, int lda,
       const float* __restrict__ B, int ldb,
       const float* __restrict__ scale, const float* __restrict__ bias,
       float* __restrict__ C, int ldc, int K,
       float* __restrict__ sumbuf, float* __restrict__ ssqbuf) {
  int w = threadIdx.x >> 5, lane = threadIdx.x & 31;
  int lane15 = lane & 15, hi = lane >> 4;
  int m0 = blockIdx.x * 32 + (w & 1) * 16;
  int n0 = blockIdx.y * 64 + (w >> 1) * 16;
  int arow = m0 + lane15;
  int bcol = n0 + lane15;
  const float* Ap = A + arow * lda + 2 * hi;
  const float* Bp = B + 2 * hi * ldb + bcol;
  v8f acc = {};
  for (int kb0 = 0; kb0 < K; kb0 += 64) {
    if (kb0 + 64 < K) {                    // stream next K-block into cache
      __builtin_prefetch(Ap + kb0 + 64, 0, 3);
      __builtin_prefetch(Bp + (kb0 + 64) * ldb, 0, 3);
    }
    #pragma unroll
    for (int kk = 0; kk < 64; kk += 4) {
      int ka = kb0 + kk;
      float2 av = *reinterpret_cast<const float2*>(Ap + ka);
      v2f a; a.x = av.x; a.y = av.y;
      float b0 = Bp[ka * ldb];
      float b1 = Bp[(ka + 1) * ldb];
      if (HAS_SCALE) { b0 *= scale[ka + 2 * hi]; b1 *= scale[ka + 2 * hi + 1]; }
      v2f bv; bv.x = b0; bv.y = b1;
      acc = wmma4(a, bv, acc);
    }
  }
  float bn = bias[bcol];
  float lsum = 0.f, lssq = 0.f;
  #pragma unroll
  for (int v = 0; v < 8; ++v) {
    float c = acc[v] + bn;
    if (RELU) c = fmaxf(c, 0.f);
    int m = m0 + 8 * hi + v;
    C[m * ldc + bcol] = c;
    if (STATS) { lsum += c; lssq += c * c; }
  }
  if (STATS) {
    atomicAdd(&sumbuf[bcol], lsum);
    atomicAdd(&ssqbuf[bcol], lssq);
  }
}

// ---------------- pairwise pass 1: rel/wgt hidden column stats ----------------
// one block per (b,i); rows j=0..63; x[j] = [states[b,i,:] | states[b,j,:]]
__global__ void __launch_bounds__(256)
k_pair_stats(const float* __restrict__ states,
             const float* __restrict__ rel_w1, const float* __restrict__ rel_b1,
             const float* __restrict__ wgt_w1, const float* __restrict__ wgt_b1,
             float* __restrict__ relsum, float* __restrict__ relssq,
             float* __restrict__ wgtsum, float* __restrict__ wgtssq) {
  extern __shared__ float smem[];
  float* xs = smem;                 // 64x256
  float* colacc = smem + 64 * 256;  // [0..255] rel sum, [256..511] wgt sum,
                                    // [512..767] rel ssq, [768..1023] wgt ssq
  int bi = blockIdx.x;
  int b = bi >> 6, i = bi & 63;
  int t = threadIdx.x;
  const float* si = states + (b * 64 + i) * 128;
  for (int idx = t; idx < 64 * 128; idx += 256) {
    int j = idx >> 7, k = idx & 127;
    xs[j * 256 + k]       = si[k];
    xs[j * 256 + 128 + k] = states[(b * 64 + j) * 128 + k];
  }
  for (int idx = t; idx < 1024; idx += 256) colacc[idx] = 0.f;
  __syncthreads();

  int w = t >> 5, lane = t & 31;
  int mt = w & 3, mat = w >> 2;     // 4 j-tiles x {rel,wgt}
  const float* W1 = mat ? wgt_w1 : rel_w1;
  const float* B1 = mat ? wgt_b1 : rel_b1;
  int lane15 = lane & 15, hi = lane >> 4;
  int arow = mt * 16 + lane15;
  for (int nt = 0; nt < 16; ++nt) {
    int n = nt * 16 + lane15;
    v8f acc = {};
    #pragma unroll 8
    for (int kb = 0; kb < 256; kb += 4) {
      int ka = kb + 2 * hi;
      v2f a; a.x = xs[arow * 256 + ka]; a.y = xs[arow * 256 + ka + 1];
      v2f bv; bv.x = W1[ka * 256 + n]; bv.y = W1[(ka + 1) * 256 + n];
      acc = wmma4(a, bv, acc);
    }
    float bias = B1[n];
    float lsum = 0.f, lssq = 0.f;
    #pragma unroll
    for (int v = 0; v < 8; ++v) {
      float h = fmaxf(acc[v] + bias, 0.f);
      lsum += h; lssq += h * h;
    }
    atomicAdd(&colacc[mat * 256 + n], lsum);
    atomicAdd(&colacc[512 + mat * 256 + n], lssq);
  }
  __syncthreads();
  atomicAdd(&relsum[t], colacc[t]);
  atomicAdd(&wgtsum[t], colacc[256 + t]);
  atomicAdd(&relssq[t], colacc[512 + t]);
  atomicAdd(&wgtssq[t], colacc[768 + t]);
}

// ---------------- pairwise pass 2: logits, gumbel weight, KL, enc_rel ----------
__global__ void __launch_bounds__(256)
k_pair_out(const float* __restrict__ states,
           const float* __restrict__ rel_w1, const float* __restrict__ rel_b1,
           const float* __restrict__ wgt_w1, const float* __restrict__ wgt_b1,
           const float* __restrict__ rel_scale, const float* __restrict__ rel_w2,
           const float* __restrict__ rel_bias2,
           const float* __restrict__ wgt_scale, const float* __restrict__ wgt_w2,
           const float* __restrict__ wgt_bias2,
           const float* __restrict__ gumbel,
           float* __restrict__ wout, float* __restrict__ kl_acc,
           float* __restrict__ xbuf) {
  extern __shared__ float smem[];
  float* xs     = smem;                       // 64x256 = 16384
  float* hrelT  = smem + 16384;               // hrelT[k*68 + j], 256x68 = 17408
  float* logits = smem + 16384 + 17408;       // 64x2
  float* wls    = logits + 128;               // 64
  float* hw     = wls + 64;                   // 256
  float* misc   = hw + 256;                   // [0]=Wsum, [1]=kl partial
  int bi = blockIdx.x;
  int b = bi >> 6, i = bi & 63;
  int t = threadIdx.x;
  const float* si = states + (b * 64 + i) * 128;
  for (int idx = t; idx < 64 * 128; idx += 256) {
    int j = idx >> 7, k = idx & 127;
    xs[j * 256 + k]       = si[k];
    xs[j * 256 + 128 + k] = states[(b * 64 + j) * 128 + k];
  }
  if (t < 128) logits[t] = 0.f;
  if (t == 0) misc[1] = 0.f;
  __syncthreads();

  int w = t >> 5, lane = t & 31;
  int mt = w & 3, mat = w >> 2;
  int lane15 = lane & 15, hi = lane >> 4;
  int arow = mt * 16 + lane15;
  if (mat == 0) {
    // rel hidden -> transposed LDS (unnormalized relu'd h)
    for (int nt = 0; nt < 16; ++nt) {
      int n = nt * 16 + lane15;
      v8f acc = {};
      #pragma unroll 8
      for (int kb = 0; kb < 256; kb += 4) {
        int ka = kb + 2 * hi;
        v2f a; a.x = xs[arow * 256 + ka]; a.y = xs[arow * 256 + ka + 1];
        v2f bv; bv.x = rel_w1[ka * 256 + n]; bv.y = rel_w1[(ka + 1) * 256 + n];
        acc = wmma4(a, bv, acc);
      }
      float bias = rel_b1[n];
      #pragma unroll
      for (int v = 0; v < 8; ++v) {
        int m = mt * 16 + 8 * hi + v;
        hrelT[n * 68 + m] = fmaxf(acc[v] + bias, 0.f);
      }
    }
  } else {
    // wgt hidden, contracted immediately with folded (scale .* w2) -> logits
    float l0[8], l1[8];
    #pragma unroll
    for (int v = 0; v < 8; ++v) { l0[v] = 0.f; l1[v] = 0.f; }
    for (int nt = 0; nt < 16; ++nt) {
      int n = nt * 16 + lane15;
      v8f acc = {};
      #pragma unroll 8
      for (int kb = 0; kb < 256; kb += 4) {
        int ka = kb + 2 * hi;
        v2f a; a.x = xs[arow * 256 + ka]; a.y = xs[arow * 256 + ka + 1];
        v2f bv; bv.x = wgt_w1[ka * 256 + n]; bv.y = wgt_w1[(ka + 1) * 256 + n];
        acc = wmma4(a, bv, acc);
      }
      float bias = wgt_b1[n];
      float s = wgt_scale[n];
      float w20 = s * wgt_w2[n * 2 + 0];
      float w21 = s * wgt_w2[n * 2 + 1];
      #pragma unroll
      for (int v = 0; v < 8; ++v) {
        float h = fmaxf(acc[v] + bias, 0.f);
        l0[v] += h * w20; l1[v] += h * w21;
      }
    }
    #pragma unroll
    for (int v = 0; v < 8; ++v) {
      int m = mt * 16 + 8 * hi + v;
      atomicAdd(&logits[m * 2 + 0], l0[v]);
      atomicAdd(&logits[m * 2 + 1], l1[v]);
    }
  }
  __syncthreads();

  if (t < 64) {
    int j = t;
    float L0 = logits[j * 2 + 0] + wgt_bias2[0];
    float L1 = logits[j * 2 + 1] + wgt_bias2[1];
    const float* g = gumbel + (bi * 64 + j) * 2;
    // straight-through gumbel: forward value is one_hot(argmax(logits+g))
    float wj = ((L1 + g[1]) > (L0 + g[0])) ? 0.f : 1.f;
    if (j == i) wj = 0.f;                      // (1 - eye) mask
    wls[j] = wj;
    wout[bi * 64 + j] = wj;
    // KL(prob || prior)
    float mx = fmaxf(L0, L1);
    float e0 = expf(L0 - mx), e1 = expf(L1 - mx);
    float inv = 1.f / (e0 + e1);
    float p0 = e0 * inv, p1 = e1 * inv;
    float kl = p0 * (logf(p0 + KL_EPS) - logf(0.05f)) +
               p1 * (logf(p1 + KL_EPS) - logf(0.95f));
    atomicAdd(&misc[1], kl);
  }
  __syncthreads();

  // contract 0/1 weights against hidden first: hw[k] = sum_j w[j]*h_rel[j,k]
  {
    float acc = 0.f;
    const float* hT = hrelT + t * 68;
    #pragma unroll 8
    for (int j = 0; j < 64; ++j) acc += wls[j] * hT[j];
    hw[t] = acc;
  }
  if (t == 0) {
    float Ws = 0.f;
    for (int j = 0; j < 64; ++j) Ws += wls[j];
    misc[0] = Ws;
    atomicAdd(kl_acc, misc[1]);
  }
  __syncthreads();

  if (t < 128) {
    float Ws = misc[0];
    float acc = Ws * rel_bias2[t];
    #pragma unroll 8
    for (int k = 0; k < 256; ++k) acc += hw[k] * rel_scale[k] * rel_w2[k * 128 + t];
    xbuf[bi * 256 + 128 + t] = acc;            // enc_rel half of x
  }
}

// ---------------- GRU elementwise ----------------
__global__ void __launch_bounds__(256)
k_gru(const float* __restrict__ gi, const float* __restrict__ gh,
      const float* __restrict__ states, float* __restrict__ upd) {
  int idx = blockIdx.x * 256 + threadIdx.x;    // 4096*128
  int row = idx >> 7, d = idx & 127;
  const float* gir = gi + row * 384;
  const float* ghr = gh + row * 384;
  float r = 1.f / (1.f + expf(-(gir[d] + ghr[d])));
  float z = 1.f / (1.f + expf(-(gir[128 + d] + ghr[128 + d])));
  float n = tanhf(gir[256 + d] + r * ghr[256 + d]);
  float s = states[idx];
  upd[idx] = (1.f - z) * n + z * s;
}

// ---------------- pred output (N=2 matvec) + KL scalar ----------------
__global__ void __launch_bounds__(256)
k_pred_out(const float* __restrict__ h, const float* __restrict__ scale,
           const float* __restrict__ w2, const float* __restrict__ bias2,
           float* __restrict__ out) {
  int idx = blockIdx.x * 256 + threadIdx.x;    // 4096*2
  int row = idx >> 1, c = idx & 1;
  float acc = bias2[c];
  const float* hr = h + row * 256;
  #pragma unroll 8
  for (int k = 0; k < 256; ++k) acc += hr[k] * scale[k] * w2[k * 2 + c];
  out[idx] = acc;
}

__global__ void k_kl(const float* __restrict__ acc, float* __restrict__ out) {
  if (threadIdx.x == 0) out[0] = acc[0] * (1.f / 4096.f);   // / (N*B)
}

// =======================================================================
extern "C" void kernel_launch(void* const* d_in, const int* in_sizes, int n_in,
                              void* d_out, int out_size, void* d_ws, size_t ws_size,
                              hipStream_t stream) {
  const float* states   = (const float*)d_in[0];
  const float* z_where  = (const float*)d_in[1];
  const float* actions  = (const float*)d_in[2];
  const float* gumbel   = (const float*)d_in[3];
  const float* self_w1  = (const float*)d_in[4];
  const float* self_b1  = (const float*)d_in[5];
  const float* self_g   = (const float*)d_in[6];
  const float* self_bb  = (const float*)d_in[7];
  const float* self_w2  = (const float*)d_in[8];
  const float* self_b2  = (const float*)d_in[9];
  const float* rel_w1   = (const float*)d_in[10];
  const float* rel_b1   = (const float*)d_in[11];
  const float* rel_g    = (const float*)d_in[12];
  const float* rel_bb   = (const float*)d_in[13];
  const float* rel_w2   = (const float*)d_in[14];
  const float* rel_b2   = (const float*)d_in[15];
  const float* wgt_w1   = (const float*)d_in[16];
  const float* wgt_b1   = (const float*)d_in[17];
  const float* wgt_g    = (const float*)d_in[18];
  const float* wgt_bb   = (const float*)d_in[19];
  const float* wgt_w2   = (const float*)d_in[20];
  const float* wgt_b2   = (const float*)d_in[21];
  const float* pred_w1  = (const float*)d_in[22];
  const float* pred_b1  = (const float*)d_in[23];
  const float* pred_g   = (const float*)d_in[24];
  const float* pred_bb  = (const float*)d_in[25];
  const float* pred_w2  = (const float*)d_in[26];
  const float* pred_b2  = (const float*)d_in[27];
  const float* gru_wih  = (const float*)d_in[28];
  const float* gru_whh  = (const float*)d_in[29];
  const float* gru_bih  = (const float*)d_in[30];
  const float* gru_bhh  = (const float*)d_in[31];

  float* out = (float*)d_out;
  float* ws  = (float*)d_ws;

  // 0) fresh stats each call (atomics accumulate)
  k_zero<<<(STATS_LEN + 255) / 256, 256, 0, stream>>>(ws, STATS_LEN);

  // 1) self hidden (one-hot collapsed) + stats
  k_self_hidden<<<256, 256, 0, stream>>>(z_where, actions, self_w1, self_b1,
                                         ws + H_SELF, ws + SELF_SUM, ws + SELF_SSQ);

  // 2) pairwise pass-1 stats (WMMA)
  k_pair_stats<<<4096, 256, (64 * 256 + 1024) * sizeof(float), stream>>>(
      states, rel_w1, rel_b1, wgt_w1, wgt_b1,
      ws + REL_SUM, ws + REL_SSQ, ws + WGT_SUM, ws + WGT_SSQ);

  // 3) BN folds
  k_finalize<<<1, 256, 0, stream>>>(ws + SELF_SUM, ws + SELF_SSQ, 4096.f,
                                    self_g, self_bb, self_w2, 128, self_b2, 128,
                                    ws + SELF_SCALE, ws + SELF_BIAS2);
  k_finalize<<<1, 256, 0, stream>>>(ws + REL_SUM, ws + REL_SSQ, 262144.f,
                                    rel_g, rel_bb, rel_w2, 128, rel_b2, 128,
                                    ws + REL_SCALE, ws + REL_BIAS2);
  k_finalize<<<1, 256, 0, stream>>>(ws + WGT_SUM, ws + WGT_SSQ, 262144.f,
                                    wgt_g, wgt_bb, wgt_w2, 2, wgt_b2, 2,
                                    ws + WGT_SCALE, ws + WGT_BIAS2);

  // 4) enc_self = h_self @ (s .* w2) + bias2  -> x[:, 0:128]  (WMMA)
  k_gemm<true, false, false><<<dim3(128, 2), 256, 0, stream>>>(
      ws + H_SELF, 256, self_w2, 128, ws + SELF_SCALE, ws + SELF_BIAS2,
      ws + XBUF, 256, 256, nullptr, nullptr);

  // 5) pairwise pass-2: logits/weight/KL + enc_rel -> x[:, 128:256]  (WMMA)
  k_pair_out<<<4096, 256,
               (16384 + 17408 + 128 + 64 + 256 + 8) * sizeof(float), stream>>>(
      states, rel_w1, rel_b1, wgt_w1, wgt_b1,
      ws + REL_SCALE, rel_w2, ws + REL_BIAS2,
      ws + WGT_SCALE, wgt_w2, ws + WGT_BIAS2,
      gumbel, out + OUT_W, ws + KL_ACC, ws + XBUF);

  // 6) GRU gates (WMMA GEMMs)
  k_gemm<false, false, false><<<dim3(128, 6), 256, 0, stream>>>(
      ws + XBUF, 256, gru_wih, 384, nullptr, gru_bih, ws + GI_BUF, 384, 256,
      nullptr, nullptr);
  k_gemm<false, false, false><<<dim3(128, 6), 256, 0, stream>>>(
      states, 128, gru_whh, 384, nullptr, gru_bhh, ws + GH_BUF, 384, 128,
      nullptr, nullptr);
  k_gru<<<2048, 256, 0, stream>>>(ws + GI_BUF, ws + GH_BUF, states, out + OUT_UPD);

  // 7) pred hidden (WMMA, relu + stats), fold, output
  k_gemm<false, true, true><<<dim3(128, 4), 256, 0, stream>>>(
      out + OUT_UPD, 128, pred_w1, 256, nullptr, pred_b1, ws + H_PRED, 256, 128,
      ws + PRED_SUM, ws + PRED_SSQ);
  k_finalize<<<1, 256, 0, stream>>>(ws + PRED_SUM, ws + PRED_SSQ, 4096.f,
                                    pred_g, pred_bb, pred_w2, 2, pred_b2, 2,
                                    ws + PRED_SCALE, ws + PRED_BIAS2);
  k_pred_out<<<32, 256, 0, stream>>>(ws + H_PRED, ws + PRED_SCALE, pred_w2,
                                     ws + PRED_BIAS2, out + OUT_ZW);
  k_kl<<<1, 32, 0, stream>>>(ws + KL_ACC, out + OUT_KL);
}